// GCN_51367808860521
// MI455X (gfx1250) — compile-verified
//
#include <hip/hip_runtime.h>

#define NN    50000
#define NE    600000
#define DIM   128
#define MROWS 64                      // rows per GEMM block = 4 tiles of 16
#define NNPAD ((NN + MROWS - 1) / MROWS * MROWS)   // 50048: h/deg padded rows

typedef __attribute__((ext_vector_type(2))) float v2f;
typedef __attribute__((ext_vector_type(8))) float v8f;

// ---------------------------------------------------------------------------
// K0: zero output + (padded) degree scratch (harness poisons buffers w/ 0xAA)
// ---------------------------------------------------------------------------
__global__ void gcn_zero_kernel(float* __restrict__ out, float* __restrict__ deg,
                                int out_n, int deg_n) {
    int i = blockIdx.x * blockDim.x + threadIdx.x;
    int stride = gridDim.x * blockDim.x;
    for (int j = i; j < out_n; j += stride) out[j] = 0.0f;
    for (int j = i; j < deg_n; j += stride) deg[j] = 0.0f;
}

// ---------------------------------------------------------------------------
// K1: sender / receiver degrees (fp32 atomics into L2)
// ---------------------------------------------------------------------------
__global__ void gcn_degree_kernel(const int* __restrict__ senders,
                                  const int* __restrict__ receivers,
                                  float* __restrict__ sdeg,
                                  float* __restrict__ rdeg) {
    int e = blockIdx.x * blockDim.x + threadIdx.x;
    if (e >= NE) return;
    unsafeAtomicAdd(&sdeg[senders[e]], 1.0f);
    unsafeAtomicAdd(&rdeg[receivers[e]], 1.0f);
}

// ---------------------------------------------------------------------------
// K1b: transpose W (128x128) -> Wt[n*DIM + k], so the WMMA B-fragment's
// K-pair per lane is contiguous in memory (enables global_load_b64).
// ---------------------------------------------------------------------------
__global__ void gcn_transpose_kernel(const float* __restrict__ W,
                                     float* __restrict__ Wt) {
    int i = blockIdx.x * blockDim.x + threadIdx.x;   // 16384 threads
    int k = i >> 7;          // row of W
    int n = i & 127;         // col of W
    Wt[(size_t)n * DIM + k] = W[(size_t)k * DIM + n];
}

// ---------------------------------------------------------------------------
// K2: h = (nodes @ W + b) * rsqrt(max(sdeg,1))    [fp32 WMMA 16x16x4]
// Block = 256 threads = 8 waves, covers 64 rows (4 M-tiles of 16).
// Wave w owns output columns [16w, 16w+16) and keeps 4 accumulators, so each
// B fragment (1 global_load_b64 from Wt) is reused by 4 v_wmma ops.
// h and sdeg are padded to NNPAD rows, so the epilogue stores UNGUARDED:
// pad rows hold garbage that no later kernel reads; padded sdeg is zeroed.
// ---------------------------------------------------------------------------
__global__ void __launch_bounds__(256)
gcn_gemm_wmma_kernel(const float* __restrict__ nodes,
                     const float* __restrict__ Wt,     // transposed weights
                     const float* __restrict__ bias,
                     const float* __restrict__ sdeg,   // NNPAD entries
                     float* __restrict__ h) {          // NNPAD x DIM
    __shared__ float ldsA[MROWS * DIM];   // 64x128 fp32 = 32 KB

    const int tid    = threadIdx.x;
    const int wave   = tid >> 5;
    const int lane   = tid & 31;
    const int m_base = blockIdx.x * MROWS;
    const int n_base = wave * 16;

    // Stage A tile (64x128 = 2048 float4) into LDS: 8 float4 per thread.
    // Clamp source row (nodes has only NN rows); keeps EXEC all-ones.
    {
        float4* dst = (float4*)ldsA;
#pragma unroll
        for (int it = 0; it < 8; ++it) {
            int f   = tid + it * 256;          // float4 index, 32 per row
            int row = f >> 5;
            int g   = m_base + row;
            if (g > NN - 1) g = NN - 1;
            dst[f] = ((const float4*)(nodes + (size_t)g * DIM))[f & 31];
        }
    }
    __syncthreads();

    // WMMA f32 16x16x4 operand layout (wave32):
    //  A 16x4: lanes 0-15 -> {K=0,K=1}, lanes 16-31 -> {K=2,K=3}, M = lane&15
    //  B 4x16: lanes 0-15 -> {K=0,K=1}, lanes 16-31 -> {K=2,K=3}, N = lane&15
    //  C/D 16x16: VGPR r -> M = r + 8*(lane>=16), N = lane&15
    const int l15  = lane & 15;
    const int hi   = lane >> 4;        // 0 or 1
    const int koff = hi * 2;

    const float* wt_lane = Wt + (size_t)(n_base + l15) * DIM + koff; // contiguous K-pair
    const float* a_lane  = ldsA + l15 * DIM + koff;

    v8f c0 = {}, c1 = {}, c2 = {}, c3 = {};
#pragma unroll 2
    for (int k = 0; k < DIM; k += 4) {
        v2f b  = *(const v2f*)(wt_lane + k);                   // global_load_b64
        v2f a0 = *(const v2f*)(a_lane + k);                    // ds b64
        v2f a1 = *(const v2f*)(a_lane + 16 * DIM + k);
        v2f a2 = *(const v2f*)(a_lane + 32 * DIM + k);
        v2f a3 = *(const v2f*)(a_lane + 48 * DIM + k);
        c0 = __builtin_amdgcn_wmma_f32_16x16x4_f32(false, a0, false, b, (short)0, c0, false, false);
        c1 = __builtin_amdgcn_wmma_f32_16x16x4_f32(false, a1, false, b, (short)0, c1, false, false);
        c2 = __builtin_amdgcn_wmma_f32_16x16x4_f32(false, a2, false, b, (short)0, c2, false, false);
        c3 = __builtin_amdgcn_wmma_f32_16x16x4_f32(false, a3, false, b, (short)0, c3, false, false);
    }

    const int   n  = n_base + l15;
    const float bn = bias[n];
    v8f acc[4] = {c0, c1, c2, c3};

    // Unguarded epilogue: h/sdeg are padded to NNPAD rows.
#pragma unroll
    for (int t = 0; t < 4; ++t) {
#pragma unroll
        for (int r = 0; r < 8; ++r) {
            const int   m  = m_base + t * 16 + r + hi * 8;
            const float rs = rsqrtf(fmaxf(sdeg[m], 1.0f));
            h[(size_t)m * DIM + n] = (acc[t][r] + bn) * rs;
        }
    }
}

// ---------------------------------------------------------------------------
// K3: edge scatter: out[r,:] += h[s,:]   (one wave per edge, float4 per lane)
// ---------------------------------------------------------------------------
__global__ void __launch_bounds__(256)
gcn_scatter_kernel(const float* __restrict__ h,
                   const int* __restrict__ senders,
                   const int* __restrict__ receivers,
                   float* __restrict__ out) {
    const int gwave = (blockIdx.x * blockDim.x + threadIdx.x) >> 5;
    const int lane  = threadIdx.x & 31;
    if (gwave >= NE) return;
    const int s = senders[gwave];
    const int r = receivers[gwave];
    const float4 v = *(const float4*)(h + (size_t)s * DIM + lane * 4);
    float* dst = out + (size_t)r * DIM + lane * 4;
    unsafeAtomicAdd(dst + 0, v.x);
    unsafeAtomicAdd(dst + 1, v.y);
    unsafeAtomicAdd(dst + 2, v.z);
    unsafeAtomicAdd(dst + 3, v.w);
}

// ---------------------------------------------------------------------------
// K4: out *= rsqrt(max(rdeg,1)) in place
// ---------------------------------------------------------------------------
__global__ void gcn_scale_kernel(float* __restrict__ out,
                                 const float* __restrict__ rdeg) {
    int i = blockIdx.x * blockDim.x + threadIdx.x;
    if (i >= NN * DIM) return;
    out[i] *= rsqrtf(fmaxf(rdeg[i >> 7], 1.0f));
}

// ---------------------------------------------------------------------------
extern "C" void kernel_launch(void* const* d_in, const int* in_sizes, int n_in,
                              void* d_out, int out_size, void* d_ws, size_t ws_size,
                              hipStream_t stream) {
    const float* nodes     = (const float*)d_in[0];
    const int*   senders   = (const int*)d_in[1];
    const int*   receivers = (const int*)d_in[2];
    const float* W         = (const float*)d_in[3];
    const float* bias      = (const float*)d_in[4];
    float*       out       = (float*)d_out;

    // Workspace: sdeg[NNPAD] | rdeg[NNPAD] | h[NNPAD*DIM] | Wt[DIM*DIM] (~26 MB)
    float* sdeg = (float*)d_ws;
    float* rdeg = sdeg + NNPAD;
    float* h    = rdeg + NNPAD;
    float* Wt   = h + (size_t)NNPAD * DIM;

    gcn_zero_kernel<<<2048, 256, 0, stream>>>(out, sdeg, NN * DIM, 2 * NNPAD);
    gcn_degree_kernel<<<(NE + 255) / 256, 256, 0, stream>>>(senders, receivers, sdeg, rdeg);
    gcn_transpose_kernel<<<(DIM * DIM) / 256, 256, 0, stream>>>(W, Wt);
    gcn_gemm_wmma_kernel<<<NNPAD / MROWS, 256, 0, stream>>>(nodes, Wt, bias, sdeg, h);
    gcn_scatter_kernel<<<(NE * 32 + 255) / 256, 256, 0, stream>>>(h, senders, receivers, out);
    gcn_scale_kernel<<<(NN * DIM + 255) / 256, 256, 0, stream>>>(out, rdeg);
}